// VolumeSDFRenderer_11106785427866
// MI455X (gfx1250) — compile-verified
//
#include <hip/hip_runtime.h>

// CDNA5 / gfx1250 NeuS volume renderer.
// cumprod -> exp(-cumsum); 16-wide inclusive prefix scan for 16 rays at once
// via V_WMMA_F32_16X16X4_F32 with a constant upper-triangular B mask.
// Memory-bound (~168 MB reads, ~7us floor @ 23.3 TB/s); all global loads are
// b128/b96 coalesced. Transcendentals use native v_exp_f32 / v_rcp_f32 so the
// VALU stream stays far under the HBM roofline.

typedef __attribute__((ext_vector_type(2))) float v2f;
typedef __attribute__((ext_vector_type(8))) float v8f;

#define NPTS 128
#define WAVES 4            // waves per block (wave32)
#define RAYS_PER_WAVE 16
#define LDS_STRIDE 132     // 132 f32: keeps rows 16B-aligned, spreads LDS banks
#define FAR_DELTA 1e10f

__device__ __forceinline__ float fast_exp(float x) {
    // native v_exp_f32 (exp2) + log2(e) scale; ~1 ulp, ample for this kernel
    return __expf(x);
}

__device__ __forceinline__ float sdf_density(float d, float s) {
    // s * e / (1+e)^2 == s * sig * (1 - sig), sig = 1/(1+exp(-d*s)).
    // v_rcp_f32: e==Inf -> sig==0 -> density 0 (no Inf/NaN into WMMA).
    float e = fast_exp(-d * s);
    float sig = __builtin_amdgcn_rcpf(1.0f + e);
    return s * sig * (1.0f - sig);
}

__global__ __launch_bounds__(32 * WAVES)
void VolumeSDFRenderer_kernel(const float* __restrict__ sl,
                              const float* __restrict__ dist,
                              const float* __restrict__ col,
                              const float* __restrict__ var,
                              float* __restrict__ out) {
    __shared__ __align__(16) float abuf[WAVES][RAYS_PER_WAVE][LDS_STRIDE];

    const int lane = threadIdx.x & 31;
    const int wv   = threadIdx.x >> 5;
    const int base = (blockIdx.x * WAVES + wv) * RAYS_PER_WAVE;

    const float v = var[0];
    const float s = fminf(fmaxf(fast_exp(v * 10.0f), 1e-6f), 1e6f);

    // ---- Phase 1: a[p] = delta[p] * density[p] into LDS (coalesced b128) ----
    for (int r = 0; r < RAYS_PER_WAVE; ++r) {
        const size_t row = (size_t)(base + r) * NPTS;
        const float4 slv = ((const float4*)(sl + row))[lane];
        const float4 dv  = ((const float4*)(dist + row))[lane];
        // neighbor lane's first depth for the cross-lane forward difference
        const float nxt = __shfl(slv.x, lane + 1, 32);
        const float d0 = slv.y - slv.x;
        const float d1 = slv.z - slv.y;
        const float d2 = slv.w - slv.z;
        const float d3 = (lane == 31) ? FAR_DELTA : (nxt - slv.w);
        float4 av;
        av.x = d0 * sdf_density(dv.x, s);
        av.y = d1 * sdf_density(dv.y, s);
        av.z = d2 * sdf_density(dv.z, s);
        av.w = d3 * sdf_density(dv.w, s);
        ((float4*)&abuf[wv][r][0])[lane] = av;
    }
    // Same-wave LDS: DS ops are in-order per wave; compiler inserts dscnt waits.

    // ---- Phase 2: per-chunk inclusive scan via WMMA f32 16x16x4 ----
    const int hf = lane >> 4;   // half-wave: A/B hold K={0,1} (lo) or K={2,3} (hi)
    const int n  = lane & 15;   // column index (scan position / C-matrix N)
    const int m  = lane & 15;   // A-matrix row M = lane & 15

    float carry[8];
    float accr[8], accg[8], accb[8];
#pragma unroll
    for (int r = 0; r < 8; ++r) { carry[r] = 0.f; accr[r] = 0.f; accg[r] = 0.f; accb[r] = 0.f; }

    for (int j = 0; j < NPTS / 16; ++j) {
        const int cbase = j * 16;
        v8f c = {0.f, 0.f, 0.f, 0.f, 0.f, 0.f, 0.f, 0.f};
#pragma unroll
        for (int t = 0; t < 4; ++t) {
            const int kk = t * 4 + hf * 2;           // K index within the chunk
            const float2 af = *(const float2*)&abuf[wv][m][cbase + kk];
            v2f A; A.x = af.x; A.y = af.y;           // a[m][cbase+kk], a[m][cbase+kk+1]
            v2f B;                                   // upper-tri ones: include k <= n
            B.x = (kk     <= n) ? 1.0f : 0.0f;
            B.y = (kk + 1 <= n) ? 1.0f : 0.0f;
            c = __builtin_amdgcn_wmma_f32_16x16x4_f32(
                    false, A, false, B, (short)0, c, false, false);
        }
        // c[r] = chunk-inclusive sum S for ray (r + 8*hf) at point (cbase + n)
#pragma unroll
        for (int r = 0; r < 8; ++r) {
            const int m2 = r + 8 * hf;
            const float S_in = carry[r] + c[r];
            const float aval = abuf[wv][m2][cbase + n];
            const float S_ex = S_in - aval;
            const float w = fast_exp(-S_ex) - fast_exp(-S_in);  // T_excl*(1-trans)
            const float* cp = col + ((size_t)(base + m2) * NPTS + (cbase + n)) * 3;
            accr[r] += w * cp[0];
            accg[r] += w * cp[1];
            accb[r] += w * cp[2];
            // carry += chunk total (column 15 of this half-wave)
            carry[r] += __shfl(c[r], (lane & 16) + 15, 32);
        }
    }

    // ---- Phase 3: reduce columns (16 lanes per half-wave) and store ----
#pragma unroll
    for (int r = 0; r < 8; ++r) {
#pragma unroll
        for (int off = 8; off >= 1; off >>= 1) {
            accr[r] += __shfl_xor(accr[r], off, 32);
            accg[r] += __shfl_xor(accg[r], off, 32);
            accb[r] += __shfl_xor(accb[r], off, 32);
        }
    }
    if (n == 0) {   // lane 0 writes rays base..base+7, lane 16 writes base+8..base+15
#pragma unroll
        for (int r = 0; r < 8; ++r) {
            const int ray = base + r + 8 * hf;
            out[(size_t)ray * 3 + 0] = accr[r];
            out[(size_t)ray * 3 + 1] = accg[r];
            out[(size_t)ray * 3 + 2] = accb[r];
        }
    }
}

extern "C" void kernel_launch(void* const* d_in, const int* in_sizes, int n_in,
                              void* d_out, int out_size, void* d_ws, size_t ws_size,
                              hipStream_t stream) {
    const float* sl   = (const float*)d_in[0];  // sample_lengths [N,128]
    const float* dist = (const float*)d_in[1];  // distance       [N,128]
    const float* col  = (const float*)d_in[2];  // color          [N,128,3]
    const float* var  = (const float*)d_in[3];  // variance scalar
    float* out = (float*)d_out;                 // [N,3]

    const int n_rays = in_sizes[1] / NPTS;      // 65536
    const int rays_per_block = WAVES * RAYS_PER_WAVE;   // 64
    const int grid = n_rays / rays_per_block;           // 1024 (N divisible)

    VolumeSDFRenderer_kernel<<<dim3(grid), dim3(32 * WAVES), 0, stream>>>(
        sl, dist, col, var, out);
}